// DEMFeatureGenerator_75453985457289
// MI455X (gfx1250) — compile-verified
//
#include <hip/hip_runtime.h>
#include <math.h>

// ---------------------------------------------------------------------------
// DEM feature generator for MI455X (gfx1250, wave32).
// Bandwidth-bound (~0.7 GB total traffic @ 23.3 TB/s => ~30us floor).
// mean15 column pass implemented as a banded matmul via V_WMMA_F32_16X16X4_F32.
// ---------------------------------------------------------------------------

typedef __attribute__((ext_vector_type(2))) float v2f;
typedef __attribute__((ext_vector_type(8))) float v8f;

#define PI_F     3.14159265358979323846f
#define INV2PI_F 0.15915494309189535f
#define HW       1024
#define IMPIX    (1024 * 1024)

// numpy/jnp 'reflect' (mirror, edge not repeated); max overhang is 7 < 1024
__device__ __forceinline__ int refl(int i, int n) {
  i = (i < 0) ? -i : i;
  return (i >= n) ? (2 * n - 2 - i) : i;
}

// Order-preserving float<->uint encoding for atomic min/max on floats.
__device__ __forceinline__ unsigned fenc(float f) {
  unsigned u = __float_as_uint(f);
  return (u & 0x80000000u) ? ~u : (u | 0x80000000u);
}
__device__ __forceinline__ float fdec(unsigned u) {
  return (u & 0x80000000u) ? __uint_as_float(u ^ 0x80000000u)
                           : __uint_as_float(~u);
}

// ---------------------------------------------------------------------------
// Kernel 0: init per-(b,ch) min/max slots (ws is poisoned; must init each call)
// mm[0..47] = min slots (encoded), mm[48..95] = max slots (encoded)
// ---------------------------------------------------------------------------
__global__ void dem_init_minmax(unsigned* __restrict__ mm) {
  if (threadIdx.x < 48) {
    mm[threadIdx.x] = 0xFFFFFFFFu;  // > enc(+inf): identity for atomicMin
    mm[48 + threadIdx.x] = 0u;      // < enc(-inf): identity for atomicMax
  }
}

// ---------------------------------------------------------------------------
// Kernel 1: fused RGB->gray (*255 folded into weights) + 3x3 Gaussian (refl).
// 32x32 output tile, 34x34 LDS tile, 256 threads.
// sigma=0.8 => 1D weights w0=0.52201140, w1=0.23899430 (normalized).
// ---------------------------------------------------------------------------
__global__ void dem_gray_gauss(const float* __restrict__ x,
                               float* __restrict__ g) {
  __shared__ float sg[34][36];
  const int b = blockIdx.z;
  const int x0 = blockIdx.x * 32, y0 = blockIdx.y * 32;
  const size_t ib = (size_t)b * 3 * IMPIX;

  for (int t = threadIdx.x; t < 34 * 34; t += 256) {
    int ly = t / 34, lx = t % 34;
    int gy = refl(y0 - 1 + ly, HW);
    int gx = refl(x0 - 1 + lx, HW);
    size_t o = ib + (size_t)gy * HW + gx;
    // 0.299/0.587/0.114 * 255 (min-max normalize makes the x255 scale exact)
    sg[ly][lx] = 76.245f * x[o] + 149.685f * x[o + IMPIX] + 29.07f * x[o + 2 * IMPIX];
  }
  __syncthreads();

  const float w0 = 0.52201140f, w1 = 0.23899430f;
  const float c00 = w1 * w1, c01 = w1 * w0, c11 = w0 * w0;
#pragma unroll
  for (int i = 0; i < 4; ++i) {
    int p = threadIdx.x + 256 * i;
    int r = p >> 5, c = p & 31;
    int yy = r + 1, xx = c + 1;
    float v = c00 * (sg[yy - 1][xx - 1] + sg[yy - 1][xx + 1] +
                     sg[yy + 1][xx - 1] + sg[yy + 1][xx + 1]) +
              c01 * (sg[yy - 1][xx] + sg[yy + 1][xx] +
                     sg[yy][xx - 1] + sg[yy][xx + 1]) +
              c11 * sg[yy][xx];
    g[(size_t)b * IMPIX + (size_t)(y0 + r) * HW + (x0 + c)] = v;
  }
}

// ---------------------------------------------------------------------------
// Kernel 2: all 6 features from blurred gray, fused min/max reduction.
// 32x32 output tile w/ halo 7 (46x46 LDS), 128 threads = 4 waves.
// mean15: separable. Row sums in LDS, then the 15-tap column sum is a banded
// 16x30 x 30x16 matmul done with 8 chained V_WMMA_F32_16X16X4_F32 per wave
// (one 16x16 subtile per wave). A[m][k] = (m<=k<=m+14)/225.
// f32 WMMA layouts (ISA 7.12.2): A/B frag = v2f, lanes 0-15 hold K={0,1},
// lanes 16-31 hold K={2,3}; M/N = lane%16. D: VGPR j -> M=j+8*(lane>=16).
// ---------------------------------------------------------------------------
__global__ void dem_features(const float* __restrict__ g,
                             float* __restrict__ out,
                             unsigned* __restrict__ mm) {
  __shared__ float sg[46][48];   // blurred gray tile + halo 7
  __shared__ float rs[48][33];   // 15-tap row sums (stride 33: no bank conflict)
  __shared__ float red[4][12];   // per-wave {min[6],max[6]}

  const int b = blockIdx.z;
  const int x0 = blockIdx.x * 32, y0 = blockIdx.y * 32;
  const size_t gb = (size_t)b * IMPIX;

  for (int t = threadIdx.x; t < 46 * 46; t += 128) {
    int ly = t / 46, lx = t % 46;
    int gy = refl(y0 - 7 + ly, HW);
    int gx = refl(x0 - 7 + lx, HW);
    sg[ly][lx] = g[gb + (size_t)gy * HW + gx];
  }
  __syncthreads();

  // 15-tap horizontal sums: rs[r][c] covers output col c, rows y0-7+r
  for (int t = threadIdx.x; t < 46 * 32; t += 128) {
    int r = t >> 5, c = t & 31;
    float s = 0.f;
#pragma unroll
    for (int k = 0; k < 15; ++k) s += sg[r][c + k];
    rs[r][c] = s;
  }
  __syncthreads();

  float mnv[6], mxv[6];
#pragma unroll
  for (int c = 0; c < 6; ++c) { mnv[c] = 3.4e38f; mxv[c] = -3.4e38f; }

  // ---- channel 5 (local_diff = g - mean15) via WMMA banded matmul ----
  const int lane = threadIdx.x & 31;
  const int wv = threadIdx.x >> 5;
  const int R = (wv >> 1) << 4;   // subtile row offset {0,16}
  const int Cc = (wv & 1) << 4;   // subtile col offset {0,16}
  const int half = lane >> 4;     // lanes 16-31 carry K+2/K+3 & M/N rows 8-15
  const int MN = lane & 15;

  v8f acc = {};
#pragma unroll
  for (int s = 0; s < 8; ++s) {   // K = 30 padded to 32, slices of 4
    int ka = 4 * s + 2 * half;
    const float w = 1.0f / 225.0f;
    v2f a, bf;
    a.x = (ka >= MN && ka <= MN + 14) ? w : 0.f;
    a.y = (ka + 1 >= MN && ka + 1 <= MN + 14) ? w : 0.f;
    bf.x = (ka < 30) ? rs[R + ka][Cc + MN] : 0.f;       // rs sized 48 rows: safe
    bf.y = (ka + 1 < 30) ? rs[R + ka + 1][Cc + MN] : 0.f;
    acc = __builtin_amdgcn_wmma_f32_16x16x4_f32(false, a, false, bf, (short)0,
                                                acc, false, false);
  }
#pragma unroll
  for (int j = 0; j < 8; ++j) {
    int ro = R + j + 8 * half;
    int co = Cc + MN;
    float ld = sg[ro + 7][co + 7] - acc[j];
    out[((size_t)(b * 6 + 5) << 20) + (size_t)(y0 + ro) * HW + (x0 + co)] = ld;
    mnv[5] = fminf(mnv[5], ld);
    mxv[5] = fmaxf(mxv[5], ld);
  }

  // ---- channels 0..4: elevation, slope, aspect, curvature, roughness ----
  const size_t cb = (size_t)(b * 6) << 20;
#pragma unroll
  for (int i = 0; i < 8; ++i) {
    int p = threadIdx.x + 128 * i;
    int ro = p >> 5, co = p & 31;
    int yy = ro + 7, xx = co + 7;
    float tl = sg[yy - 1][xx - 1], tc = sg[yy - 1][xx], tr = sg[yy - 1][xx + 1];
    float ml = sg[yy][xx - 1],     mc = sg[yy][xx],     mr = sg[yy][xx + 1];
    float bl = sg[yy + 1][xx - 1], bc = sg[yy + 1][xx], br = sg[yy + 1][xx + 1];
    // cross-correlation (torch/jax conv semantics)
    float gx = (tr - tl) + 2.f * (mr - ml) + (br - bl);
    float gy = (bl - tl) + 2.f * (bc - tc) + (br - tr);
    float slope = sqrtf(gx * gx + gy * gy);
    float aspect = (atan2f(gy, gx) + PI_F) * INV2PI_F;
    float curv = 2.f * (tl + tr + bl + br) - 8.f * mc;
    float s = 0.f, s2 = 0.f;
#pragma unroll
    for (int dy = -2; dy <= 2; ++dy)
#pragma unroll
      for (int dx = -2; dx <= 2; ++dx) {
        float v = sg[yy + dy][xx + dx];
        s += v; s2 += v * v;
      }
    float m = s * 0.04f, m2 = s2 * 0.04f;
    float rough = sqrtf(fmaxf(m2 - m * m, 0.f));

    size_t po = (size_t)(y0 + ro) * HW + (x0 + co);
    out[cb + (0ull << 20) + po] = mc;
    out[cb + (1ull << 20) + po] = slope;
    out[cb + (2ull << 20) + po] = aspect;
    out[cb + (3ull << 20) + po] = curv;
    out[cb + (4ull << 20) + po] = rough;
    mnv[0] = fminf(mnv[0], mc);     mxv[0] = fmaxf(mxv[0], mc);
    mnv[1] = fminf(mnv[1], slope);  mxv[1] = fmaxf(mxv[1], slope);
    mnv[2] = fminf(mnv[2], aspect); mxv[2] = fmaxf(mxv[2], aspect);
    mnv[3] = fminf(mnv[3], curv);   mxv[3] = fmaxf(mxv[3], curv);
    mnv[4] = fminf(mnv[4], rough);  mxv[4] = fmaxf(mxv[4], rough);
  }

  // ---- fused block min/max reduction (wave32 shuffles) + global atomics ----
#pragma unroll
  for (int off = 16; off > 0; off >>= 1) {
#pragma unroll
    for (int c = 0; c < 6; ++c) {
      mnv[c] = fminf(mnv[c], __shfl_xor(mnv[c], off, 32));
      mxv[c] = fmaxf(mxv[c], __shfl_xor(mxv[c], off, 32));
    }
  }
  if (lane == 0) {
#pragma unroll
    for (int c = 0; c < 6; ++c) { red[wv][c] = mnv[c]; red[wv][6 + c] = mxv[c]; }
  }
  __syncthreads();
  if (threadIdx.x < 12) {
    int c = threadIdx.x;
    float a0 = red[0][c], a1 = red[1][c], a2 = red[2][c], a3 = red[3][c];
    if (c < 6) {
      atomicMin(&mm[b * 6 + c], fenc(fminf(fminf(a0, a1), fminf(a2, a3))));
    } else {
      atomicMax(&mm[48 + b * 6 + (c - 6)],
                fenc(fmaxf(fmaxf(a0, a1), fmaxf(a2, a3))));
    }
  }
}

// ---------------------------------------------------------------------------
// Kernel 3: in-place per-(b,ch) min-max normalize of d_out, float4 vectorized.
// Each slot covers 2^20 floats = 2^18 float4s -> slot = i >> 18.
// ---------------------------------------------------------------------------
__global__ void dem_normalize(float* __restrict__ out,
                              const unsigned* __restrict__ mm, int total4) {
  int i = blockIdx.x * blockDim.x + threadIdx.x;
  if (i >= total4) return;
  int slot = i >> 18;
  float mn = fdec(mm[slot]);
  float mx = fdec(mm[48 + slot]);
  float inv = 1.f / fmaxf(mx - mn, 1e-8f);
  float4 f = reinterpret_cast<float4*>(out)[i];
  f.x = (f.x - mn) * inv;
  f.y = (f.y - mn) * inv;
  f.z = (f.z - mn) * inv;
  f.w = (f.w - mn) * inv;
  reinterpret_cast<float4*>(out)[i] = f;
}

// ---------------------------------------------------------------------------
// Workspace layout: [0,384) min/max uints (96), [1024, 1024+32MB) blurred gray
// ---------------------------------------------------------------------------
extern "C" void kernel_launch(void* const* d_in, const int* in_sizes, int n_in,
                              void* d_out, int out_size, void* d_ws,
                              size_t ws_size, hipStream_t stream) {
  const float* x = (const float*)d_in[0];
  float* out = (float*)d_out;
  unsigned* mm = (unsigned*)d_ws;
  float* gbuf = (float*)((char*)d_ws + 1024);

  dem_init_minmax<<<1, 64, 0, stream>>>(mm);
  dem_gray_gauss<<<dim3(32, 32, 8), 256, 0, stream>>>(x, gbuf);
  dem_features<<<dim3(32, 32, 8), 128, 0, stream>>>(gbuf, out, mm);
  const int total4 = (8 * 6) << 18;  // 12,582,912 float4s
  dem_normalize<<<total4 / 256, 256, 0, stream>>>(out, mm, total4);
}